// SpatialTransformer_91302414778602
// MI455X (gfx1250) — compile-verified
//
#include <hip/hip_runtime.h>
#include <hip/hip_bf16.h>
#include <math.h>

#define KCP 20            // control points
#define NM  23            // K+3
#define OH  64
#define OW  256
#define NP  (OH*OW)       // 16384 grid points
#define IH  128
#define IW  512
#define NB  128

typedef __attribute__((ext_vector_type(2))) float v2f;
typedef __attribute__((ext_vector_type(8))) float v8f;

__device__ __forceinline__ float ctrl_x(int k) {
    int i = (k < 10) ? k : (k - 10);
    return 0.05f + 0.1f * (float)i;
}
__device__ __forceinline__ float ctrl_y(int k) {
    return (k < 10) ? 0.05f : 0.95f;
}

// ---------------------------------------------------------------------------
// Kernel 1: build TPS delta matrix and invert (Gauss-Jordan, partial pivot,
// double precision to match numpy's float64 inverse). One block.
// ---------------------------------------------------------------------------
__global__ __launch_bounds__(256) void build_inv_delta(float* __restrict__ invd) {
    __shared__ double M[NM][2 * NM];   // augmented [delta | I]
    __shared__ double fac[NM];
    __shared__ int pivrow;
    const int t = threadIdx.x;

    for (int e = t; e < NM * 2 * NM; e += 256) {
        int i = e / (2 * NM), j = e % (2 * NM);
        double v;
        if (j >= NM) {
            v = (j - NM == i) ? 1.0 : 0.0;
        } else if (i < KCP) {
            if (j == 0)      v = 1.0;
            else if (j == 1) v = (double)ctrl_x(i);
            else if (j == 2) v = (double)ctrl_y(i);
            else {
                int k = j - 3;
                if (k == i) v = 0.0;   // fill_diagonal(1) -> 1^2*log(1)=0
                else {
                    double dx = (double)ctrl_x(i) - (double)ctrl_x(k);
                    double dy = (double)ctrl_y(i) - (double)ctrl_y(k);
                    double r2 = dx * dx + dy * dy;
                    v = r2 * log(sqrt(r2));
                }
            }
        } else if (i < KCP + 2) {
            v = (j < 3) ? 0.0 : ((i == KCP) ? (double)ctrl_x(j - 3)
                                            : (double)ctrl_y(j - 3));
        } else {
            v = (j < 3) ? 0.0 : 1.0;
        }
        M[i][j] = v;
    }
    __syncthreads();

    for (int p = 0; p < NM; ++p) {
        if (t == 0) {                       // partial pivot search (serial, 23 elems)
            int best = p; double bv = fabs(M[p][p]);
            for (int r = p + 1; r < NM; ++r) {
                double av = fabs(M[r][p]);
                if (av > bv) { bv = av; best = r; }
            }
            pivrow = best;
        }
        __syncthreads();
        int pr = pivrow;
        if (pr != p) {
            for (int c = t; c < 2 * NM; c += 256) {
                double tmp = M[p][c]; M[p][c] = M[pr][c]; M[pr][c] = tmp;
            }
        }
        __syncthreads();
        double pv = M[p][p];
        __syncthreads();                    // all read pivot before row writes
        for (int c = t; c < 2 * NM; c += 256) M[p][c] = M[p][c] / pv;
        __syncthreads();
        if (t < NM) fac[t] = (t == p) ? 0.0 : M[t][p];
        __syncthreads();
        for (int e = t; e < NM * 2 * NM; e += 256) {
            int r = e / (2 * NM), c = e % (2 * NM);
            if (r != p) M[r][c] -= fac[r] * M[p][c];
        }
        __syncthreads();
    }

    for (int e = t; e < NM * NM; e += 256)
        invd[e] = (float)M[e / NM][NM + e % NM];
}

// ---------------------------------------------------------------------------
// Kernel 2: T = INVD[23x23] @ Cp0[23x256]  (columns = (batch,coord) pairs)
// via v_wmma_f32_16x16x4_f32.  M: 2 tiles (32), N: 16 tiles (256), K: 6 steps.
// One workgroup of 32 waves; wave w -> tile (mt = w&1, nt = w>>1).
// ---------------------------------------------------------------------------
__global__ __launch_bounds__(1024) void compute_T_wmma(
        const float* __restrict__ invd,    // [23*23]
        const float* __restrict__ ctrl,    // [128,20,2]
        float* __restrict__ Tws) {         // [23][256]
    const int lane = threadIdx.x & 31;
    const int wave = threadIdx.x >> 5;
    const int mt = wave & 1;
    const int nt = wave >> 1;
    const int halfsel = lane >> 4;         // lanes 16-31 hold K+2 / M+8 halves
    const int l16 = lane & 15;

    const int row  = mt * 16 + l16;        // A row (M)
    const int coln = nt * 16 + l16;        // B/D column (N) = b*2 + c
    const int bb = coln >> 1;
    const int cc = coln & 1;

    v8f acc = {};
    #pragma unroll
    for (int kk = 0; kk < 6; ++kk) {
        int k0 = kk * 4 + halfsel * 2;
        v2f a, b;
        // A fragment: 16x4 slice of INVD (zero-padded to 32x24)
        a.x = (row < NM && (k0)     < NM) ? invd[row * NM + k0]     : 0.0f;
        a.y = (row < NM && (k0 + 1) < NM) ? invd[row * NM + k0 + 1] : 0.0f;
        // B fragment: 4x16 slice of Cp0 (rows 20..23 are zeros)
        b.x = ((k0)     < KCP) ? ctrl[bb * (KCP * 2) + (k0)     * 2 + cc] : 0.0f;
        b.y = ((k0 + 1) < KCP) ? ctrl[bb * (KCP * 2) + (k0 + 1) * 2 + cc] : 0.0f;
        acc = __builtin_amdgcn_wmma_f32_16x16x4_f32(
                  false, a, false, b, (short)0, acc, false, false);
    }
    #pragma unroll
    for (int r = 0; r < 8; ++r) {
        int m = mt * 16 + r + halfsel * 8;
        if (m < NM) Tws[m * 256 + coln] = acc[r];
    }
}

// ---------------------------------------------------------------------------
// Kernel 3: precompute lifted grid G (transposed [23][16384] for coalescing).
// Done once per launch; amortizes 20 sqrt+log per pixel across 128 batches.
// ---------------------------------------------------------------------------
__global__ __launch_bounds__(256) void precompute_G(float* __restrict__ Gws) {
    int n = blockIdx.x * 256 + threadIdx.x;
    if (n >= NP) return;
    int oy = n >> 8, ox = n & 255;
    float gx = ((float)ox + 0.5f) * (1.0f / (float)OW);
    float gy = ((float)oy + 0.5f) * (1.0f / (float)OH);
    Gws[0 * NP + n] = 1.0f;
    Gws[1 * NP + n] = gx;
    Gws[2 * NP + n] = gy;
    #pragma unroll
    for (int k = 0; k < KCP; ++k) {
        float dx = gx - ctrl_x(k);
        float dy = gy - ctrl_y(k);
        float d2 = dx * dx + dy * dy;
        float d  = sqrtf(d2);
        Gws[(3 + k) * NP + n] = d2 * logf(d + 1e-6f);
    }
}

// ---------------------------------------------------------------------------
// Kernel 4 (hot, HBM-bound): per block: 256 pixels x 16 batches.
// G row stays in registers across the batch loop; T slice in LDS.
// ---------------------------------------------------------------------------
__global__ __launch_bounds__(256) void sample_kernel(
        const float* __restrict__ X,       // [128,128,512,3]
        const float* __restrict__ Tws,     // [23][256]
        const float* __restrict__ Gws,     // [23][16384]
        float* __restrict__ out) {         // [128,64,256,3]
    __shared__ float Tl[NM][32];           // [i][bl*2+c] for this batch group
    const int t = threadIdx.x;
    const int tile = blockIdx.x;           // 0..63 pixel tiles
    const int bg = blockIdx.y;             // 0..7 batch groups of 16

    for (int e = t; e < NM * 32; e += 256) {
        int i = e >> 5, col = e & 31;
        Tl[i][col] = Tws[i * 256 + bg * 32 + col];
    }
    __syncthreads();

    const int n = tile * 256 + t;          // this thread's output pixel
    float g[NM];
    #pragma unroll
    for (int i = 0; i < NM; ++i) g[i] = Gws[i * NP + n];
    const int oy = n >> 8, ox = n & 255;

    #pragma unroll 2
    for (int bl = 0; bl < 16; ++bl) {
        const int b = bg * 16 + bl;
        float Gx = 0.0f, Gy = 0.0f;
        #pragma unroll
        for (int i = 0; i < NM; ++i) {
            Gx = fmaf(g[i], Tl[i][bl * 2 + 0], Gx);
            Gy = fmaf(g[i], Tl[i][bl * 2 + 1], Gy);
        }
        float fx = fminf(fmaxf((float)IW * Gx, 0.0f), (float)(IW - 2));
        float fy = fminf(fmaxf((float)IH * Gy, 0.0f), (float)(IH - 2));
        float fx0 = floorf(fx), fy0 = floorf(fy);
        int x0 = (int)fx0, y0 = (int)fy0;
        float ax = fx - fx0, ay = fy - fy0;

        const float* p00 = X + (((size_t)b * IH + y0) * IW + x0) * 3;
        const float* p01 = p00 + (size_t)IW * 3;       // (y0+1, x0)
        float i00r = p00[0], i00g = p00[1], i00b = p00[2];
        float i10r = p00[3], i10g = p00[4], i10b = p00[5];   // (y0, x0+1)
        float i01r = p01[0], i01g = p01[1], i01b = p01[2];
        float i11r = p01[3], i11g = p01[4], i11b = p01[5];   // (y0+1, x0+1)

        float w00 = (1.0f - ax) * (1.0f - ay);
        float w10 = ax * (1.0f - ay);
        float w01 = (1.0f - ax) * ay;
        float w11 = ax * ay;

        float r = w00 * i00r + w01 * i01r + w10 * i10r + w11 * i11r;
        float gch = w00 * i00g + w01 * i01g + w10 * i10g + w11 * i11g;
        float bch = w00 * i00b + w01 * i01b + w10 * i10b + w11 * i11b;

        float* o = out + (((size_t)b * OH + oy) * OW + ox) * 3;
        o[0] = r; o[1] = gch; o[2] = bch;
    }
}

// ---------------------------------------------------------------------------
extern "C" void kernel_launch(void* const* d_in, const int* in_sizes, int n_in,
                              void* d_out, int out_size, void* d_ws, size_t ws_size,
                              hipStream_t stream) {
    (void)in_sizes; (void)n_in; (void)out_size; (void)ws_size;
    const float* X    = (const float*)d_in[0];   // [128,128,512,3]
    const float* ctrl = (const float*)d_in[1];   // [128,20,2]
    float* out = (float*)d_out;

    float* ws = (float*)d_ws;
    float* ws_invd = ws;                 // 529 floats
    float* ws_T    = ws + 576;           // 23*256 = 5888 floats
    float* ws_G    = ws + 576 + 5888;    // 23*16384 = 376832 floats (~1.5 MB)

    build_inv_delta<<<1, 256, 0, stream>>>(ws_invd);
    compute_T_wmma<<<1, 1024, 0, stream>>>(ws_invd, ctrl, ws_T);
    precompute_G<<<NP / 256, 256, 0, stream>>>(ws_G);
    sample_kernel<<<dim3(64, 8), 256, 0, stream>>>(X, ws_T, ws_G, out);
}